// RPGaussianPooling_27857157882435
// MI455X (gfx1250) — compile-verified
//
#include <hip/hip_runtime.h>
#include <math.h>

// RPGaussianPooling fused kernel for gfx1250 (MI455X).
// Memory-bound (~268MB min traffic): single fused pass, f32 WMMA for projections,
// non-temporal hints on the streaming load/store paths (footprint > 192MB L2).

typedef float v2f __attribute__((ext_vector_type(2)));
typedef float v8f __attribute__((ext_vector_type(8)));

#define C_DIM 64
#define NB 8
#define NRANK 4
#define T_LEN 8192
#define B_SZ 64
#define PROJ 32                       // NRANK * NB
#define WAVES_PER_WG 8
#define T_PER_WAVE 16
#define T_PER_WG (WAVES_PER_WG * T_PER_WAVE)   // 128

__global__ __launch_bounds__(256)
void rp_gauss_pool_kernel(const float* __restrict__ x,
                          const float* __restrict__ sigma,
                          const float* __restrict__ rho,
                          const float* __restrict__ E,
                          const float* __restrict__ F,
                          float* __restrict__ out)
{
    // Scaled transposed weights: rows 0..31 = E-proj (p=r*8+n), rows 32..63 = F-proj.
    __shared__ float Wt[2 * PROJ][C_DIM];   // 16 KB

    const int tid  = threadIdx.x;
    const int wg   = blockIdx.x;
    const int b    = wg / (T_LEN / T_PER_WG);
    const int tblk = (wg % (T_LEN / T_PER_WG)) * T_PER_WG;

    // ---- Phase 0: cooperative scaled weight load, transposed into LDS ----
    // E/F/sigma/rho are re-read by every block: leave them regular-temporal so
    // they stay L2-resident; only x/out stream with NT hints.
    const float scale = 8.0f;  // sqrt(IN_DIM=64)
    for (int i = tid; i < 2 * PROJ * C_DIM; i += 256) {
        const int row = i >> 6;      // 0..63
        const int c   = i & 63;
        float val;
        if (row < PROJ) {
            const int r = row >> 3, n = row & 7;
            const float s = scale / (1e-5f + fabsf(sigma[r]));
            val = s * E[(r * C_DIM + c) * NB + n];
        } else {
            const int p = row - PROJ;
            const int r = p >> 3, n = p & 7;
            const float s = scale / (1e-5f + fabsf(rho[r]));
            val = s * F[(r * C_DIM + c) * NB + n];
        }
        Wt[row][c] = val;
    }
    __syncthreads();

    // ---- Phase 1: per-wave GEMM over a 16-wide t tile ----
    const int wave  = tid >> 5;
    const int lane  = tid & 31;
    const int ln    = lane & 15;     // matrix row (A) / t column (B,D)
    const int khalf = lane >> 4;     // which K pair this half-wave holds

    const int t0 = tblk + wave * T_PER_WAVE;
    const float* xb = x + (size_t)b * C_DIM * T_LEN;

    v8f accElo = {}; v8f accEhi = {}; v8f accFlo = {}; v8f accFhi = {};

#pragma unroll
    for (int K = 0; K < C_DIM / 4; ++K) {
        const int c0 = 4 * K + 2 * khalf;

        // B fragment: B[k][n] = x[b, 4K+k, t0+n]
        // lanes 0-15: v0=K0,v1=K1 ; lanes 16-31: v0=K2,v1=K3
        // x is read exactly once per launch -> non-temporal loads.
        const float* xp = xb + (size_t)c0 * T_LEN + t0 + ln;
        v2f bfrag;
        bfrag.x = __builtin_nontemporal_load(xp);
        bfrag.y = __builtin_nontemporal_load(xp + T_LEN);

        // A fragments: A[m][k] = Wt[rowBase+m][4K+k] -> one aligned 8B LDS read each
        const v2f aElo = *(const v2f*)&Wt[     ln][c0];
        const v2f aEhi = *(const v2f*)&Wt[16 + ln][c0];
        const v2f aFlo = *(const v2f*)&Wt[32 + ln][c0];
        const v2f aFhi = *(const v2f*)&Wt[48 + ln][c0];

        accElo = __builtin_amdgcn_wmma_f32_16x16x4_f32(false, aElo, false, bfrag,
                                                       (short)0, accElo, false, false);
        accEhi = __builtin_amdgcn_wmma_f32_16x16x4_f32(false, aEhi, false, bfrag,
                                                       (short)0, accEhi, false, false);
        accFlo = __builtin_amdgcn_wmma_f32_16x16x4_f32(false, aFlo, false, bfrag,
                                                       (short)0, accFlo, false, false);
        accFhi = __builtin_amdgcn_wmma_f32_16x16x4_f32(false, aFhi, false, bfrag,
                                                       (short)0, accFhi, false, false);
    }

    // ---- Phase 2: epilogue ----
    // D layout: lane holds column t = t0 + ln; rows M = v + 8*khalf.
    // accElo -> e of rank r = khalf     (n = 0..7)
    // accEhi -> e of rank r = khalf + 2 (n = 0..7)
    // Partner lane (lane ^ 16) holds the SAME t with the other two ranks.
    float eA[8], eB[8], fA[8], fB[8];
    float eAr[8], eBr[8], fAr[8], fBr[8];
#pragma unroll
    for (int v = 0; v < 8; ++v) {
        eA[v] = accElo[v]; eB[v] = accEhi[v];
        fA[v] = accFlo[v]; fB[v] = accFhi[v];
    }
#pragma unroll
    for (int v = 0; v < 8; ++v) {
        eAr[v] = __shfl_xor(eA[v], 16, 32);
        eBr[v] = __shfl_xor(eB[v], 16, 32);
        fAr[v] = __shfl_xor(fA[v], 16, 32);
        fBr[v] = __shfl_xor(fB[v], 16, 32);
    }

    // Each half-wave computes its 32 of the 64 outputs j = n*8 + m:
    // lanes 0-15 -> n in [0,4), lanes 16-31 -> n in [4,8)
    const int nbase = 4 * khalf;
    float o[32];
    float mx = 0.0f;
#pragma unroll
    for (int i = 0; i < 4; ++i) {
        const int n = nbase + i;
#pragma unroll
        for (int m = 0; m < 8; ++m) {
            float z = eA[n] * fA[m] + eB[n] * fB[m]
                    + eAr[n] * fAr[m] + eBr[n] * fBr[m];
            z *= 0.25f;  // / N_RANK
            const float so = copysignf(sqrtf(fabsf(z) + 0.01f) - 0.1f, z);
            o[i * 8 + m] = so;
            mx = fmaxf(mx, fabsf(so));
        }
    }
    // combine max across the two half-waves (same t column)
    mx = fmaxf(mx, __shfl_xor(mx, 16, 32));
    const float inv = 1.0f / (mx + 1e-5f);

    // out[b][j][t]; per j the 16-lane group writes 16 consecutive t -> coalesced.
    // out is written once and never re-read -> non-temporal stores.
    float* ob = out + (size_t)b * (NB * NB) * T_LEN + t0 + ln;
#pragma unroll
    for (int i = 0; i < 4; ++i) {
#pragma unroll
        for (int m = 0; m < 8; ++m) {
            const int j = (nbase + i) * 8 + m;
            __builtin_nontemporal_store(o[i * 8 + m] * inv, &ob[(size_t)j * T_LEN]);
        }
    }
}

extern "C" void kernel_launch(void* const* d_in, const int* in_sizes, int n_in,
                              void* d_out, int out_size, void* d_ws, size_t ws_size,
                              hipStream_t stream) {
    (void)in_sizes; (void)n_in; (void)out_size; (void)d_ws; (void)ws_size;
    const float* x     = (const float*)d_in[0];
    const float* sigma = (const float*)d_in[1];
    const float* rho   = (const float*)d_in[2];
    const float* E     = (const float*)d_in[3];
    const float* F     = (const float*)d_in[4];
    float* out = (float*)d_out;

    const int blocks = B_SZ * (T_LEN / T_PER_WG);  // 64 * 64 = 4096
    rp_gauss_pool_kernel<<<blocks, 256, 0, stream>>>(x, sigma, rho, E, F, out);
}